// Model_20658792694701
// MI455X (gfx1250) — compile-verified
//
#include <hip/hip_runtime.h>
#include <hip/hip_bf16.h>

// Problem constants (from setup_inputs): B=4, H=1024, W=1280.
#define B_IMG 4
#define H_IMG 1024
#define W_IMG 1280
#define NPIX  (H_IMG * W_IMG)

// Tiling: 128x16 output tile per 256-thread block (8 wave32).
#define TILE_X 128
#define TILE_Y 16
#define HALO_W (TILE_X + 2)      // 130
#define LDSH   (TILE_Y + 2)      // 18
#define LDSW   132               // padded row stride in LDS (floats)

// ---------------------------------------------------------------------------
// CDNA5 async global->LDS copy (ASYNCcnt path). The builtin's params are
// addrspace(1) int* (prints as "__device__ int *") and addrspace(3) int*
// (prints as "__shared__ int *") per the round-1 diagnostic.
// ---------------------------------------------------------------------------
#if defined(__has_builtin)
#if __has_builtin(__builtin_amdgcn_global_load_async_to_lds_b32)
#define HAVE_ASYNC_LDS 1
#endif
#if __has_builtin(__builtin_amdgcn_s_wait_asynccnt)
#define HAVE_WAIT_ASYNC 1
#endif
#endif

typedef __attribute__((address_space(1))) int gint_t;
typedef __attribute__((address_space(3))) int lint_t;

__device__ __forceinline__ void async_copy_f32(const float* g, float* l) {
#if defined(HAVE_ASYNC_LDS)
  __builtin_amdgcn_global_load_async_to_lds_b32((gint_t*)g, (lint_t*)l, 0, 0);
#else
  *l = *g;  // synchronous fallback
#endif
}

__device__ __forceinline__ void wait_async0() {
#if defined(HAVE_ASYNC_LDS)
#if defined(HAVE_WAIT_ASYNC)
  __builtin_amdgcn_s_wait_asynccnt(0);
#else
  asm volatile("s_wait_asynccnt 0" ::: "memory");
#endif
#endif
}

// ---------------------------------------------------------------------------
// One Jacobi propagation step:
//   dst = sum_t k_t(y,x) * src(y + i_t, x + j_t),  zero-padded borders,
// where k is recomputed from kernel_raw each iteration (cheaper than storing
// the 189MB 9-channel tensor; bandwidth is ~identical, VALU is free).
// OFFSETS channel map (value read = d[y+i][x+j]):
//   ch0:(-1,-1) ch1:(-1,0) ch2:(-1,+1) ch3:(0,+1)
//   ch4:(+1,+1) ch5:(+1,0) ch6:(+1,-1) ch7:(0,-1)   center weight = 1 - sum
// ---------------------------------------------------------------------------
__global__ __launch_bounds__(256) void cspn_step_kernel(
    const float* __restrict__ src,   // [B,1,H,W]  (current disparity)
    const float* __restrict__ kraw,  // [B,8,H,W]  (raw affinity)
    float* __restrict__ dst,         // [B,1,H,W]
    int do_clamp) {
  __shared__ float tile[LDSH * LDSW];

  const int tx0 = blockIdx.x * TILE_X;
  const int ty0 = blockIdx.y * TILE_Y;
  const int b   = blockIdx.z;
  const int tid = threadIdx.x;

  const float* sb = src + (size_t)b * NPIX;

  const bool interior = (blockIdx.x > 0) && (blockIdx.x + 1 < gridDim.x) &&
                        (blockIdx.y > 0) && (blockIdx.y + 1 < gridDim.y);

  const int NL = LDSH * HALO_W;  // 2340 halo elements
  if (interior) {
    // Async DMA of the (TILE_Y+2)x(TILE_X+2) halo tile into LDS.
    for (int i = tid; i < NL; i += 256) {
      const int ly = i / HALO_W;
      const int lx = i - ly * HALO_W;
      const float* g = sb + (size_t)(ty0 - 1 + ly) * W_IMG + (tx0 - 1 + lx);
      async_copy_f32(g, &tile[ly * LDSW + lx]);
    }
    wait_async0();
  } else {
    // Border tiles: bounds-checked synchronous staging (implements zero pad).
    for (int i = tid; i < NL; i += 256) {
      const int ly = i / HALO_W;
      const int lx = i - ly * HALO_W;
      const int gy = ty0 - 1 + ly;
      const int gx = tx0 - 1 + lx;
      float v = 0.0f;
      if (gy >= 0 && gy < H_IMG && gx >= 0 && gx < W_IMG)
        v = sb[(size_t)gy * W_IMG + gx];
      tile[ly * LDSW + lx] = v;
    }
  }
  __syncthreads();

  const float* kb = kraw + (size_t)b * 8 * NPIX;

#pragma unroll
  for (int s = 0; s < (TILE_X * TILE_Y) / 256; ++s) {
    const int p  = tid + s * 256;
    const int yl = p >> 7;            // p / TILE_X
    const int xl = p & (TILE_X - 1);  // p % TILE_X
    const int gy = ty0 + yl;
    const int gx = tx0 + xl;
    const size_t pix = (size_t)gy * W_IMG + gx;

    // Load 8 raw channels (coalesced) and normalize on the fly.
    float kr[8];
    float asum = 0.0f;
#pragma unroll
    for (int c = 0; c < 8; ++c) {
      kr[c] = kb[(size_t)c * NPIX + pix];
      asum += fabsf(kr[c]);
    }
    const float inv = 1.0f / (asum + 1e-9f);
    float ssum = 0.0f;
#pragma unroll
    for (int c = 0; c < 8; ++c) {
      kr[c] *= inv;
      ssum += kr[c];
    }

    const float* c0 = &tile[(yl + 1) * LDSW + (xl + 1)];
    float acc = (1.0f - ssum) * c0[0];
    acc = fmaf(kr[0], c0[-LDSW - 1], acc);  // NW
    acc = fmaf(kr[1], c0[-LDSW],     acc);  // N
    acc = fmaf(kr[2], c0[-LDSW + 1], acc);  // NE
    acc = fmaf(kr[3], c0[1],         acc);  // E
    acc = fmaf(kr[4], c0[LDSW + 1],  acc);  // SE
    acc = fmaf(kr[5], c0[LDSW],      acc);  // S
    acc = fmaf(kr[6], c0[LDSW - 1],  acc);  // SW
    acc = fmaf(kr[7], c0[-1],        acc);  // W

    if (do_clamp) acc = fminf(fmaxf(acc, 0.0f), 256.0f);
    dst[(size_t)b * NPIX + pix] = acc;
  }
}

// ---------------------------------------------------------------------------
// Host launcher: 24 Jacobi steps ping-ponging through d_ws (needs 2*B*H*W
// floats = 41.9 MB); last step clamps into d_out. Stream-only, graph-safe.
// ---------------------------------------------------------------------------
extern "C" void kernel_launch(void* const* d_in, const int* in_sizes, int n_in,
                              void* d_out, int out_size, void* d_ws,
                              size_t ws_size, hipStream_t stream) {
  (void)in_sizes; (void)n_in; (void)out_size; (void)ws_size;

  const float* disp = (const float*)d_in[0];   // [4,1,1024,1280]
  const float* kraw = (const float*)d_in[1];   // [4,8,1024,1280]
  float* out = (float*)d_out;

  float* bufA = (float*)d_ws;
  float* bufB = bufA + (size_t)B_IMG * NPIX;

  dim3 grid(W_IMG / TILE_X, H_IMG / TILE_Y, B_IMG);  // (10, 64, 4)
  dim3 block(256, 1, 1);

  const float* src = disp;
  for (int it = 0; it < 24; ++it) {
    const bool last = (it == 23);
    float* dst = last ? out : ((it & 1) ? bufB : bufA);
    cspn_step_kernel<<<grid, block, 0, stream>>>(src, kraw, dst, last ? 1 : 0);
    src = dst;
  }
}